// GCN_encoder_scatter_43593918054554
// MI455X (gfx1250) — compile-verified
//
#include <hip/hip_runtime.h>
#include <hip/hip_bf16.h>

typedef __attribute__((ext_vector_type(2))) float v2f;
typedef __attribute__((ext_vector_type(8))) float v8f;

#define KDIM 256      // NUM_FEATURES
#define NH   128      // HIDDEN
#define LDA  260      // padded LDS stride (float) -> conflict-free b64 reads

// ---------------- degree / weight kernels ----------------

__global__ void k_init_deg(int* __restrict__ deg, int n) {
    int i = blockIdx.x * blockDim.x + threadIdx.x;
    if (i < n) deg[i] = 1;  // self loop
}

__global__ void k_count_deg(const int* __restrict__ col, int* __restrict__ deg, int e) {
    int i = blockIdx.x * blockDim.x + threadIdx.x;
    if (i < e) atomicAdd(&deg[col[i]], 1);
}

__global__ void k_dinv(const int* __restrict__ deg, float* __restrict__ dinv, int n) {
    int i = blockIdx.x * blockDim.x + threadIdx.x;
    if (i < n) dinv[i] = rsqrtf((float)deg[i]);
}

// ---------------- exclusive scan of in-edge counts (deg-1) ----------------

__global__ void k_block_sums(const int* __restrict__ deg, int* __restrict__ bsum, int n) {
    __shared__ int sh[1024];
    int t = threadIdx.x;
    int i = blockIdx.x * 1024 + t;
    sh[t] = (i < n) ? (deg[i] - 1) : 0;
    __syncthreads();
    for (int s = 512; s > 0; s >>= 1) {
        if (t < s) sh[t] += sh[t + s];
        __syncthreads();
    }
    if (t == 0) bsum[blockIdx.x] = sh[0];
}

__global__ void k_scan_bsums(int* __restrict__ bsum, int nb) {
    __shared__ int sh[1024];
    int t = threadIdx.x;
    int v = (t < nb) ? bsum[t] : 0;
    sh[t] = v;
    __syncthreads();
    for (int off = 1; off < 1024; off <<= 1) {
        int u = (t >= off) ? sh[t - off] : 0;
        __syncthreads();
        sh[t] += u;
        __syncthreads();
    }
    if (t < nb) bsum[t] = sh[t] - v;  // exclusive
}

__global__ void k_scan_final(const int* __restrict__ deg, const int* __restrict__ bsum,
                             int* __restrict__ offsets, int* __restrict__ cursor, int n) {
    __shared__ int sh[1024];
    int t = threadIdx.x;
    int i = blockIdx.x * 1024 + t;
    int v = (i < n) ? (deg[i] - 1) : 0;
    sh[t] = v;
    __syncthreads();
    for (int off = 1; off < 1024; off <<= 1) {
        int u = (t >= off) ? sh[t - off] : 0;
        __syncthreads();
        sh[t] += u;
        __syncthreads();
    }
    if (i < n) {
        int ex = bsum[blockIdx.x] + sh[t] - v;
        offsets[i] = ex;
        cursor[i]  = ex;
    }
}

// ---------------- CSR placement: pack {src_row, weight} per edge ----------------

__global__ void k_place(const int* __restrict__ row, const int* __restrict__ col,
                        const float* __restrict__ dinv, int* __restrict__ cursor,
                        int2* __restrict__ rec, int e) {
    int i = blockIdx.x * blockDim.x + threadIdx.x;
    if (i >= e) return;
    int r = row[i], c = col[i];
    int p = atomicAdd(&cursor[c], 1);
    int2 v;
    v.x = r;
    v.y = __float_as_int(dinv[r] * dinv[c]);
    rec[p] = v;
}

// ---------------- h = x @ W via V_WMMA_F32_16X16X4_F32 ----------------
// block = 256 threads (8 waves), owns a 32-row M tile of x staged in LDS
// (stride 260 floats -> conflict-free b64 A-fragment reads for both sub-tiles).
// Each wave owns a 16-column N stripe and accumulates TWO 16x16 tiles off the
// same B fragment, halving W traffic per wmma and doubling wmma density.

__global__ void k_gemm_wmma(const float* __restrict__ x, const float* __restrict__ W,
                            float* __restrict__ h, int n) {
    __shared__ float Asl[32 * LDA];
    int m0 = blockIdx.x * 32;
    int t  = threadIdx.x;

    // cooperative load of the 32x256 A tile as float4s (2048 float4, 8/thread)
    #pragma unroll
    for (int q = 0; q < 8; ++q) {
        int f   = t + 256 * q;
        int row = f >> 6;           // 64 float4 per row
        int cv  = f & 63;
        float4 v = make_float4(0.f, 0.f, 0.f, 0.f);
        int m = m0 + row;
        if (m < n) v = *((const float4*)(x + (size_t)m * KDIM + cv * 4));
        *((float4*)(Asl + row * LDA + cv * 4)) = v;
    }
    __syncthreads();

    int wave = t >> 5, lane = t & 31;
    int n0  = wave * 16;
    int l15 = lane & 15;
    int hi  = lane >> 4;            // 0: K pair (0,1); 1: K pair (2,3)

    const float* a0p = Asl + l15 * LDA + 2 * hi;
    const float* a1p = a0p + 16 * LDA;

    v8f c0 = {}, c1 = {};
    #pragma unroll 4
    for (int k = 0; k < KDIM; k += 4) {
        v2f a0, a1, b;
        a0.x = a0p[k];     a0.y = a0p[k + 1];
        a1.x = a1p[k];     a1.y = a1p[k + 1];
        const float* bp = W + (size_t)(k + 2 * hi) * NH + n0 + l15;
        b.x = bp[0];
        b.y = bp[NH];
        c0 = __builtin_amdgcn_wmma_f32_16x16x4_f32(false, a0, false, b,
                                                   (short)0, c0, false, false);
        c1 = __builtin_amdgcn_wmma_f32_16x16x4_f32(false, a1, false, b,
                                                   (short)0, c1, false, false);
    }

    // D layout: VGPR r -> rows M=r (lanes 0-15) and M=8+r (lanes 16-31)
    int col  = n0 + l15;
    int base = m0 + 8 * hi;
    if (m0 + 31 < n) {
        // full tile: branchless, clause-able stores
        #pragma unroll
        for (int r = 0; r < 8; ++r) {
            h[(size_t)(base + r) * NH + col]      = c0[r];
            h[(size_t)(base + 16 + r) * NH + col] = c1[r];
        }
    } else {
        for (int r = 0; r < 8; ++r) {
            int m = base + r;
            if (m < n)      h[(size_t)m * NH + col]        = c0[r];
            if (m + 16 < n) h[(size_t)(m + 16) * NH + col] = c1[r];
        }
    }
}

// ---------------- pull-style aggregation: one wave per node ----------------
// 32 lanes x float4 = 128 floats per row; each in-edge is one coherent 512B burst.

__global__ void k_aggregate(const float* __restrict__ h, const int2* __restrict__ rec,
                            const int* __restrict__ offsets, const int* __restrict__ deg,
                            const float* __restrict__ dinv, const float* __restrict__ bias,
                            float* __restrict__ out, int n) {
    int wave = threadIdx.x >> 5;
    int lane = threadIdx.x & 31;
    int node = blockIdx.x * (blockDim.x >> 5) + wave;
    if (node >= n) return;

    const float4* h4 = (const float4*)h;    // 32 float4 per row
    float ds = dinv[node];
    float ws = ds * ds;                     // self-loop weight
    float4 hs = h4[(size_t)node * 32 + lane];
    float4 acc;
    acc.x = ws * hs.x; acc.y = ws * hs.y; acc.z = ws * hs.z; acc.w = ws * hs.w;

    int start = offsets[node];
    int cnt   = deg[node] - 1;
    for (int j = 0; j < cnt; ++j) {
        int2  e = rec[start + j];           // one b64 load: {row, weight}
        float w = __int_as_float(e.y);
        if (j + 1 < cnt) {
            int rn = rec[start + j + 1].x;
            __builtin_prefetch(h + (size_t)rn * NH + lane * 4, 0, 0);
        }
        float4 hv = h4[(size_t)e.x * 32 + lane];
        acc.x += w * hv.x; acc.y += w * hv.y; acc.z += w * hv.z; acc.w += w * hv.w;
    }

    const float4* b4 = (const float4*)bias;
    float4 bb = b4[lane];
    float4 o;
    o.x = acc.x + bb.x; o.y = acc.y + bb.y; o.z = acc.z + bb.z; o.w = acc.w + bb.w;
    ((float4*)out)[(size_t)node * 32 + lane] = o;
}

// ---------------- launcher ----------------

extern "C" void kernel_launch(void* const* d_in, const int* in_sizes, int n_in,
                              void* d_out, int out_size, void* d_ws, size_t ws_size,
                              hipStream_t stream) {
    const float* x     = (const float*)d_in[0];
    const int*   edges = (const int*)d_in[1];
    const float* W     = (const float*)d_in[2];
    const float* bias  = (const float*)d_in[3];
    float*       out   = (float*)d_out;

    const int N = in_sizes[0] / KDIM;    // nodes
    const int E = in_sizes[1] / 2;       // edges
    const int* erow = edges;
    const int* ecol = edges + E;

    // carve workspace (256B aligned chunks)
    char* base = (char*)d_ws;
    size_t off = 0;
    auto carve = [&](size_t bytes) -> char* {
        char* p = base + off;
        off += (bytes + 255) & ~(size_t)255;
        return p;
    };
    int*   deg     = (int*)  carve((size_t)N * 4);
    float* dinv    = (float*)carve((size_t)N * 4);
    int*   offsets = (int*)  carve((size_t)N * 4);
    int*   cursor  = (int*)  carve((size_t)N * 4);
    int*   bsum    = (int*)  carve(2048 * 4);
    int2*  recbuf  = (int2*) carve((size_t)E * 8);
    float* h       = (float*)carve((size_t)N * NH * 4);
    (void)ws_size;

    const int nb = (N + 1023) / 1024;

    k_init_deg <<<(N + 255) / 256, 256, 0, stream>>>(deg, N);
    k_count_deg<<<(E + 255) / 256, 256, 0, stream>>>(ecol, deg, E);
    k_dinv     <<<(N + 255) / 256, 256, 0, stream>>>(deg, dinv, N);

    k_block_sums<<<nb, 1024, 0, stream>>>(deg, bsum, N);
    k_scan_bsums<<<1, 1024, 0, stream>>>(bsum, nb);
    k_scan_final<<<nb, 1024, 0, stream>>>(deg, bsum, offsets, cursor, N);

    k_place<<<(E + 255) / 256, 256, 0, stream>>>(erow, ecol, dinv, cursor, recbuf, E);

    k_gemm_wmma<<<(N + 31) / 32, 256, 0, stream>>>(x, W, h, N);

    k_aggregate<<<(N + 7) / 8, 256, 0, stream>>>(h, recbuf, offsets, deg, dinv,
                                                 bias, out, N);
}